// MemoryController_72602127171964
// MI455X (gfx1250) — compile-verified
//
#include <hip/hip_runtime.h>
#include <math.h>

typedef float v2f __attribute__((ext_vector_type(2)));
typedef float v8f __attribute__((ext_vector_type(8)));

#define NWAVES   16
#define BW       16      // batches per workgroup
#define WDIM     128
#define RH       4
#define NCELL    128
#define TSTEPS   256
#define BTOT     32
#define IFACE_N  919
#define INSZ_K   640
#define DNC_DELTA 1e-6f

// LDS strides (floats), padded even + bank-staggered
#define XS   644
#define GS   932
#define HS   132
#define RVS  516

#define OFF_X   0
#define OFF_G   (OFF_X + BW*XS)
#define OFF_HE  (OFF_G + BW*GS)
#define OFF_CE  (OFF_HE + BW*HS)
#define OFF_HC  (OFF_CE + BW*HS)
#define OFF_CC  (OFF_HC + BW*HS)
#define OFF_RV  (OFF_CC + BW*HS)
#define OFF_US  (OFF_RV + BW*RVS)
#define OFF_WW  (OFF_US + BW*HS)
#define OFF_PR  (OFF_WW + BW*HS)
#define OFF_RW  (OFF_PR + BW*HS)
#define LDS_FLOATS (OFF_RW + BW*4*HS)   // ~56704 floats = ~222 KB < 320 KB/WGP

__device__ __forceinline__ float sigf(float x){ return 1.f/(1.f+expf(-x)); }
__device__ __forceinline__ float softplusf(float x){
  return (x > 0.f) ? x + log1pf(expf(-x)) : log1pf(expf(x));
}
__device__ __forceinline__ float wredsum(float v){
#pragma unroll
  for (int s = 16; s > 0; s >>= 1) v += __shfl_xor(v, s, 32);
  return v;
}
__device__ __forceinline__ float wredmax(float v){
#pragma unroll
  for (int s = 16; s > 0; s >>= 1) v = fmaxf(v, __shfl_xor(v, s, 32));
  return v;
}

// D[16 x Nout] = concat(A1[16xK1], A2[16xK2]) * [W1|W2]^T + b1 + b2
// A1/A2 in LDS (row strides lda1/lda2). W1:[Nout x K1], W2:[Nout x K2] row-major in global.
// Result to LDS (outL,ldo) or directly to global (gOut,gldo).
// Uses V_WMMA_F32_16X16X4_F32; 16 waves stripe the n-tiles.
__device__ void wmma_fc(const float* __restrict__ A1, int lda1, int K1,
                        const float* __restrict__ A2, int lda2, int K2,
                        const float* __restrict__ W1, const float* __restrict__ W2,
                        const float* __restrict__ b1, const float* __restrict__ b2,
                        int Nout, float* __restrict__ outL, int ldo,
                        float* __restrict__ gOut, long long gldo)
{
  const int lane = threadIdx.x & 31;
  const int wave = threadIdx.x >> 5;
  const int hh   = lane >> 4;      // half-wave select
  const int l16  = lane & 15;
  const int K    = K1 + K2;
  const int ntiles = (Nout + 15) >> 4;

  for (int tile = wave; tile < ntiles; tile += NWAVES) {
    const int ncol = tile*16 + l16;
    const bool nok = (ncol < Nout);
    // prefetch next tile's weight rows (L2-resident weights -> keep warm)
    {
      int nn = (tile + NWAVES)*16 + l16;
      if (nn < Nout) {
        __builtin_prefetch(&W1[(size_t)nn*(size_t)K1], 0, 1);
        if (K2 > 0) __builtin_prefetch(&W2[(size_t)nn*(size_t)K2], 0, 1);
      }
    }
    v8f acc = {};
    for (int k0 = 0; k0 < K; k0 += 4) {
      const int kk = k0 + (hh << 1);  // A 16x4 frag: lane half picks K pair (ISA 7.12.2)
      v2f a, b;
      if (kk < K1) {
        a.x = A1[l16*lda1 + kk];  a.y = A1[l16*lda1 + kk + 1];
      } else {
        const int k2 = kk - K1;
        a.x = A2[l16*lda2 + k2];  a.y = A2[l16*lda2 + k2 + 1];
      }
      if (nok) {
        if (kk < K1) { const float* wr = W1 + (size_t)ncol*K1; b.x = wr[kk];   b.y = wr[kk+1]; }
        else         { const float* wr = W2 + (size_t)ncol*K2; const int k2 = kk-K1; b.x = wr[k2]; b.y = wr[k2+1]; }
      } else { b.x = 0.f; b.y = 0.f; }
      acc = __builtin_amdgcn_wmma_f32_16x16x4_f32(false, a, false, b, (short)0, acc, false, false);
    }
    if (nok) {
      const float bias = (b1 ? b1[ncol] : 0.f) + (b2 ? b2[ncol] : 0.f);
#pragma unroll
      for (int j = 0; j < 8; ++j) {            // C/D: row = j + 8*half, col = lane&15
        const int row = j + hh*8;
        const float v = acc[j] + bias;
        if (gOut) gOut[(long long)row*gldo + ncol] = v;
        else      outL[row*ldo + ncol] = v;
      }
    }
  }
}

__device__ __forceinline__ void lstm_update(const float* __restrict__ g,
                                            float* __restrict__ h,
                                            float* __restrict__ c, int lane)
{
#pragma unroll
  for (int j = 0; j < 4; ++j) {
    const int w = lane + 32*j;
    const float ig = sigf(g[w]);
    const float fg = sigf(g[128 + w]);
    const float gg = tanhf(g[256 + w]);
    const float og = sigf(g[384 + w]);
    const float cn = fg*c[w] + ig*gg;
    c[w] = cn;
    h[w] = og*tanhf(cn);
  }
}

extern __shared__ float smem[];

__global__ __launch_bounds__(512, 1)
void dnc_persistent_kernel(const float* __restrict__ input,
                           const float* __restrict__ eWih, const float* __restrict__ eWhh,
                           const float* __restrict__ ebih, const float* __restrict__ ebhh,
                           const float* __restrict__ cWih, const float* __restrict__ cWhh,
                           const float* __restrict__ cbih, const float* __restrict__ cbhh,
                           const float* __restrict__ ifW,  const float* __restrict__ ifb,
                           const float* __restrict__ oW,   const float* __restrict__ ob,
                           float* __restrict__ out,
                           float* __restrict__ memG, float* __restrict__ linkG)
{
  float* xbuf = smem + OFF_X;    // [BW][XS]  controller/out input rows (A-matrix)
  float* gbuf = smem + OFF_G;    // [BW][GS]  gates / interface vector xi
  float* hE   = smem + OFF_HE;   // [BW][HS]
  float* cE   = smem + OFF_CE;
  float* hC   = smem + OFF_HC;
  float* cC   = smem + OFF_CC;
  float* rvB  = smem + OFF_RV;   // [BW][RVS] read vectors [w][r] (also rc-logit scratch)
  float* usg  = smem + OFF_US;   // [BW][HS]
  float* wwv  = smem + OFF_WW;
  float* prc  = smem + OFF_PR;
  float* rwv  = smem + OFF_RW;   // [BW][RH][HS]

  const int tid  = threadIdx.x;
  const int lane = tid & 31;
  const int wave = tid >> 5;
  const int boff = blockIdx.x * BW;
  const int m    = wave;          // local batch handled by this wave (wave-local phases)
  const int bg   = boff + m;

  float* memB = memG  + (size_t)bg * NCELL * WDIM;
  float* lnkB = linkG + (size_t)bg * NCELL * NCELL;

  for (int i = tid; i < LDS_FLOATS; i += blockDim.x) smem[i] = 0.f;
  __syncthreads();

  for (int t = 0; t < TSTEPS; ++t) {
    // ---- 1. load encoder input x_t -------------------------------------
#pragma unroll
    for (int j = 0; j < 4; ++j) {
      const int w = lane + 32*j;
      xbuf[m*XS + w] = input[((size_t)bg*TSTEPS + t)*WDIM + w];
    }
    __syncthreads();
    // ---- 2. encoder gates = [x,hE] @ [Wih|Whh]^T + b --------------------
    wmma_fc(xbuf, XS, WDIM, hE, HS, WDIM, eWih, eWhh, ebih, ebhh,
            4*WDIM, gbuf, GS, nullptr, 0);
    __syncthreads();
    // ---- 3. encoder LSTM (wave-local) ----------------------------------
    lstm_update(gbuf + m*GS, hE + m*HS, cE + m*HS, lane);
    // ---- 4. controller input x = interleave(enc, rv) -------------------
#pragma unroll
    for (int j = 0; j < 4; ++j) {
      const int w = lane + 32*j;
      xbuf[m*XS + w*5] = hE[m*HS + w];
#pragma unroll
      for (int r = 0; r < RH; ++r)
        xbuf[m*XS + w*5 + 1 + r] = rvB[m*RVS + w*RH + r];
    }
    __syncthreads();
    // ---- 5/6. controller LSTM ------------------------------------------
    wmma_fc(xbuf, XS, INSZ_K, hC, HS, WDIM, cWih, cWhh, cbih, cbhh,
            4*WDIM, gbuf, GS, nullptr, 0);
    __syncthreads();
    lstm_update(gbuf + m*GS, hC + m*HS, cC + m*HS, lane);
    __syncthreads();
    // ---- 7. interface vector xi = hC @ iface_W^T + b -------------------
    wmma_fc(hC, HS, WDIM, nullptr, 0, 0, ifW, nullptr, ifb, nullptr,
            IFACE_N, gbuf, GS, nullptr, 0);
    __syncthreads();

    // ---- 8. DNC memory step: one wave per batch ------------------------
    {
      float* xi    = gbuf + m*GS;
      float* keysn = xbuf + m*XS;          // [5][128] normalized keys (xbuf free now)
      // wave-uniform scalars (redundant per lane; LDS broadcast reads)
      float rstr[RH], fgate[RH], md0[RH], md1[RH], md2[RH];
#pragma unroll
      for (int r = 0; r < RH; ++r) rstr[r]  = softplusf(xi[512 + r]);
      const float wstr  = softplusf(xi[644]);
#pragma unroll
      for (int r = 0; r < RH; ++r) fgate[r] = sigf(xi[901 + r]);
      const float agate = sigf(xi[905]);
      const float wgate = sigf(xi[906]);
#pragma unroll
      for (int r = 0; r < RH; ++r) {
        const float a0 = xi[907 + r*3], a1 = xi[908 + r*3], a2 = xi[909 + r*3];
        const float mx = fmaxf(a0, fmaxf(a1, a2));
        const float e0 = expf(a0-mx), e1 = expf(a1-mx), e2 = expf(a2-mx);
        const float is = 1.f/(e0+e1+e2);
        md0[r] = e0*is; md1[r] = e1*is; md2[r] = e2*is;
      }
      // normalized keys (4 read keys + write key), norm via wave reduction
#pragma unroll
      for (int kI = 0; kI < 5; ++kI) {
        const float* src = (kI < 4) ? (xi + kI*128) : (xi + 516);
        float tv[4], ss = 0.f;
#pragma unroll
        for (int j = 0; j < 4; ++j) { const int w = lane+32*j; tv[j] = tanhf(src[w]); ss += tv[j]*tv[j]; }
        const float inv = 1.f/(sqrtf(wredsum(ss)) + DNC_DELTA);
#pragma unroll
        for (int j = 0; j < 4; ++j) keysn[kI*128 + (lane+32*j)] = tv[j]*inv;
      }
      // erase/write_vec -> scratch overlapping consumed xi[0:256)
      float* ers  = xi + 0;
      float* wvec = xi + 128;
      float* wcb  = xi + 256;
      float* alc  = xi + 384;
#pragma unroll
      for (int j = 0; j < 4; ++j) {
        const int w = lane + 32*j;
        const float e = sigf(xi[645 + w]);
        const float v = tanhf(xi[773 + w]);
        ers[w] = e; wvec[w] = v;
      }
      // usage update (uses previous ww, rw)
#pragma unroll
      for (int j = 0; j < 4; ++j) {
        const int n = lane + 32*j;
        float u0 = usg[m*HS + n];
        u0 = u0 + (1.f - u0)*wwv[m*HS + n];
        float p = 1.f;
#pragma unroll
        for (int r = 0; r < RH; ++r) p *= (1.f - fgate[r]*rwv[m*4*HS + r*HS + n]);
        usg[m*HS + n] = u0*p;
      }
      __syncthreads();   // keys/erase/usage cross-lane handoff (uniform)

      // pass1: write-content logits on pre-write memory
#pragma unroll
      for (int j = 0; j < 4; ++j) {
        const int n = lane + 32*j;
        const float* mr = memB + (size_t)n*WDIM;
        float nn = 0.f, dt = 0.f;
        for (int w = 0; w < WDIM; ++w) { const float mv = mr[w]; nn = fmaf(mv,mv,nn); dt = fmaf(mv, keysn[4*128+w], dt); }
        wcb[n] = dt/(sqrtf(nn)+DNC_DELTA) * wstr;
      }
      // softmax wc
      {
        float vals[4], mx = -INFINITY;
#pragma unroll
        for (int j = 0; j < 4; ++j) { vals[j] = wcb[lane+32*j]; mx = fmaxf(mx, vals[j]); }
        mx = wredmax(mx);
        float s = 0.f;
#pragma unroll
        for (int j = 0; j < 4; ++j) { vals[j] = expf(vals[j]-mx); s += vals[j]; }
        const float is = 1.f/wredsum(s);
#pragma unroll
        for (int j = 0; j < 4; ++j) wcb[lane+32*j] = vals[j]*is;
      }
      // allocation: stable-sort equivalent, O(N^2) product of smaller-usage cells
      {
        float uj[4];
#pragma unroll
        for (int j = 0; j < 4; ++j) uj[j] = DNC_DELTA + (1.f-DNC_DELTA)*usg[m*HS + lane+32*j];
        float p[4] = {1.f,1.f,1.f,1.f};
        for (int k = 0; k < NCELL; ++k) {
          const float uk = DNC_DELTA + (1.f-DNC_DELTA)*usg[m*HS + k];
#pragma unroll
          for (int j = 0; j < 4; ++j) {
            const int n = lane + 32*j;
            const bool before = (uk < uj[j]) || (uk == uj[j] && k < n);
            p[j] = before ? p[j]*uk : p[j];
          }
        }
#pragma unroll
        for (int j = 0; j < 4; ++j) alc[lane+32*j] = (1.f - uj[j])*p[j];
      }
      // write weighting
#pragma unroll
      for (int j = 0; j < 4; ++j) {
        const int n = lane + 32*j;
        wwv[m*HS + n] = wgate*(agate*alc[n] + (1.f-agate)*wcb[n]);
      }
      __syncthreads();   // ww cross-lane handoff (uniform)

      // pass2: erase+write memory, accumulate new norms and read-key dots
      float* rcl = rvB + m*RVS;   // [4][128] rc logits (old rv already consumed)
#pragma unroll
      for (int j = 0; j < 4; ++j) {
        const int n = lane + 32*j;
        const float wwn = wwv[m*HS + n];
        float* mr = memB + (size_t)n*WDIM;
        float nn = 0.f, d0 = 0.f, d1 = 0.f, d2 = 0.f, d3 = 0.f;
        for (int w = 0; w < WDIM; ++w) {
          float mv = mr[w];
          mv = mv*(1.f - wwn*ers[w]) + wwn*wvec[w];
          mr[w] = mv;
          nn = fmaf(mv,mv,nn);
          d0 = fmaf(mv, keysn[0*128+w], d0);
          d1 = fmaf(mv, keysn[1*128+w], d1);
          d2 = fmaf(mv, keysn[2*128+w], d2);
          d3 = fmaf(mv, keysn[3*128+w], d3);
        }
        const float inv = 1.f/(sqrtf(nn)+DNC_DELTA);
        rcl[0*128+n] = d0*inv*rstr[0];
        rcl[1*128+n] = d1*inv*rstr[1];
        rcl[2*128+n] = d2*inv*rstr[2];
        rcl[3*128+n] = d3*inv*rstr[3];
      }
      // softmax rc per read head, scale by content mode
#pragma unroll
      for (int r = 0; r < RH; ++r) {
        float vals[4], mx = -INFINITY;
#pragma unroll
        for (int j = 0; j < 4; ++j) { vals[j] = rcl[r*128 + lane+32*j]; mx = fmaxf(mx, vals[j]); }
        mx = wredmax(mx);
        float s = 0.f;
#pragma unroll
        for (int j = 0; j < 4; ++j) { vals[j] = expf(vals[j]-mx); s += vals[j]; }
        const float is = md2[r]/wredsum(s);
#pragma unroll
        for (int j = 0; j < 4; ++j) rcl[r*128 + lane+32*j] = vals[j]*is;
      }
      // link update (rows outer, lanes=cols, coalesced) + fused backward einsum
      {
        float bwa[RH][4];
#pragma unroll
        for (int r = 0; r < RH; ++r)
#pragma unroll
          for (int j = 0; j < 4; ++j) bwa[r][j] = 0.f;
        for (int n2 = 0; n2 < NCELL; ++n2) {
          const float wwn = wwv[m*HS + n2];
          float rwp[RH];
#pragma unroll
          for (int r = 0; r < RH; ++r) rwp[r] = rwv[m*4*HS + r*HS + n2];
          float* lr = lnkB + (size_t)n2*NCELL;
#pragma unroll
          for (int j = 0; j < 4; ++j) {
            const int i2 = lane + 32*j;
            float ln = (1.f - wwn - wwv[m*HS + i2])*lr[i2] + wwn*prc[m*HS + i2];
            if (i2 == n2) ln = 0.f;
            lr[i2] = ln;
#pragma unroll
            for (int r = 0; r < RH; ++r) bwa[r][j] = fmaf(rwp[r], ln, bwa[r][j]);
          }
        }
#pragma unroll
        for (int r = 0; r < RH; ++r)
#pragma unroll
          for (int j = 0; j < 4; ++j) rcl[r*128 + lane+32*j] += md0[r]*bwa[r][j];
      }
      // forward einsum: fw[r][n] = sum_j link[n][j]*rw_prev[r][j]
#pragma unroll
      for (int j = 0; j < 4; ++j) {
        const int n = lane + 32*j;
        const float* lr = lnkB + (size_t)n*NCELL;
        float fa[RH] = {0.f,0.f,0.f,0.f};
        for (int k = 0; k < NCELL; ++k) {
          const float lv = lr[k];
#pragma unroll
          for (int r = 0; r < RH; ++r) fa[r] = fmaf(lv, rwv[m*4*HS + r*HS + k], fa[r]);
        }
#pragma unroll
        for (int r = 0; r < RH; ++r) rcl[r*128 + n] += md1[r]*fa[r];
      }
      // precedence update
      {
        float wl[4], s = 0.f;
#pragma unroll
        for (int j = 0; j < 4; ++j) { wl[j] = wwv[m*HS + lane+32*j]; s += wl[j]; }
        s = wredsum(s);
#pragma unroll
        for (int j = 0; j < 4; ++j) {
          const int n = lane + 32*j;
          prc[m*HS + n] = (1.f - s)*prc[m*HS + n] + wl[j];
        }
      }
      // commit new read weights
#pragma unroll
      for (int r = 0; r < RH; ++r)
#pragma unroll
        for (int j = 0; j < 4; ++j)
          rwv[m*4*HS + r*HS + lane+32*j] = rcl[r*128 + lane+32*j];
      __syncthreads();   // rw cross-lane handoff before read_vecs (uniform)

      // pass3: read vectors rv[w][r] = sum_n rw[r][n]*mem[n][w] (coalesced in w)
      {
        float rva[4][RH];
#pragma unroll
        for (int j = 0; j < 4; ++j)
#pragma unroll
          for (int r = 0; r < RH; ++r) rva[j][r] = 0.f;
        for (int n2 = 0; n2 < NCELL; ++n2) {
          float rwn[RH];
#pragma unroll
          for (int r = 0; r < RH; ++r) rwn[r] = rwv[m*4*HS + r*HS + n2];
          const float* mr = memB + (size_t)n2*WDIM;
#pragma unroll
          for (int j = 0; j < 4; ++j) {
            const float mv = mr[lane + 32*j];
#pragma unroll
            for (int r = 0; r < RH; ++r) rva[j][r] = fmaf(rwn[r], mv, rva[j][r]);
          }
        }
#pragma unroll
        for (int j = 0; j < 4; ++j)
#pragma unroll
          for (int r = 0; r < RH; ++r)
            rvB[m*RVS + (lane+32*j)*RH + r] = rva[j][r];
      }
    }

    // ---- 9. output input = [hC, rv_flat] -------------------------------
#pragma unroll
    for (int j = 0; j < 4; ++j) {
      const int w = lane + 32*j;
      xbuf[m*XS + w] = hC[m*HS + w];
#pragma unroll
      for (int r = 0; r < RH; ++r)
        xbuf[m*XS + WDIM + w*RH + r] = rvB[m*RVS + w*RH + r];
    }
    __syncthreads();
    // ---- 10. out_t = cat @ out_W^T + b -> global [B,T,W] ----------------
    wmma_fc(xbuf, XS, INSZ_K, nullptr, 0, 0, oW, nullptr, ob, nullptr,
            WDIM, nullptr, 0,
            out + ((size_t)boff*TSTEPS + t)*WDIM, (long long)TSTEPS*WDIM);
    __syncthreads();
  }
}

extern "C" void kernel_launch(void* const* d_in, const int* in_sizes, int n_in,
                              void* d_out, int out_size, void* d_ws, size_t ws_size,
                              hipStream_t stream) {
  (void)in_sizes; (void)n_in; (void)out_size; (void)ws_size;
  const float* input = (const float*)d_in[0];
  // d_in[1] source_lengths: all == T, unused
  const float* eWih = (const float*)d_in[2];
  const float* eWhh = (const float*)d_in[3];
  const float* ebih = (const float*)d_in[4];
  const float* ebhh = (const float*)d_in[5];
  const float* cWih = (const float*)d_in[6];
  const float* cWhh = (const float*)d_in[7];
  const float* cbih = (const float*)d_in[8];
  const float* cbhh = (const float*)d_in[9];
  const float* ifW  = (const float*)d_in[10];
  const float* ifb  = (const float*)d_in[11];
  const float* oW   = (const float*)d_in[12];
  const float* ob   = (const float*)d_in[13];

  float* memG  = (float*)d_ws;                             // [B][N][W]
  float* linkG = memG + (size_t)BTOT*NCELL*WDIM;           // [B][N][N]
  const size_t zbytes = ((size_t)BTOT*NCELL*WDIM + (size_t)BTOT*NCELL*NCELL) * sizeof(float);
  hipMemsetAsync(d_ws, 0, zbytes, stream);                 // state starts at zero each call

  const size_t shmem = (size_t)LDS_FLOATS * sizeof(float); // ~222 KB dynamic LDS
  dnc_persistent_kernel<<<dim3(BTOT/BW), dim3(NWAVES*32), shmem, stream>>>(
      input, eWih, eWhh, ebih, ebhh, cWih, cWhh, cbih, cbhh,
      ifW, ifb, oW, ob, (float*)d_out, memG, linkG);
}